// GRU_LSTM_67714454389232
// MI455X (gfx1250) — compile-verified
//
#include <hip/hip_runtime.h>
#include <cmath>

typedef __attribute__((ext_vector_type(2))) float v2f;
typedef __attribute__((ext_vector_type(8))) float v8f;

// Problem constants (match reference: H=18, B=T=1024)
constexpr int kH  = 18;
constexpr int kH3 = 54;    // 3H (GRU gates)
constexpr int kH4 = 72;    // 4H (LSTM gates)
constexpr int kT  = 1024;
constexpr int kB  = 1024;
constexpr int kKP = 20;    // K padded to multiple of 4
constexpr int kKS = 5;     // K steps of 4
constexpr int kNG = 64;    // 3H padded to 4 N-tiles
constexpr int kNL = 80;    // 4H padded to 5 N-tiles

__device__ __forceinline__ float sigmoidf_(float v) {
    return 1.0f / (1.0f + __expf(-v));
}

__global__ __launch_bounds__(32) void fused_gru_lstm_fc(
    const float* __restrict__ x,
    const float* __restrict__ gWih, const float* __restrict__ gWhh,
    const float* __restrict__ gbih, const float* __restrict__ gbhh,
    const float* __restrict__ lWih, const float* __restrict__ lWhh,
    const float* __restrict__ lbih, const float* __restrict__ lbhh,
    const float* __restrict__ fcW,  const float* __restrict__ fcb,
    float* __restrict__ out)
{
    // Weights (zero-padded) staged once; B-fragments then hoisted to registers.
    __shared__ float WgX[kNG][kKP], WgH[kNG][kKP];
    __shared__ float WlX[kNL][kKP], WlH[kNL][kKP];
    __shared__ float gbi_s[kH3], gbh_s[kH3], lb_s[kH4];
    __shared__ float fcw_s[kH];
    __shared__ float fcb_s;
    // Per-tile streaming state
    __shared__ float xt[16][kKP];          // x_t tile (K-padded with zeros)
    __shared__ float hg[16][kKP];          // GRU hidden state == g_t after update
    __shared__ float hl[16][kKP];          // LSTM hidden state
    __shared__ float cs[16][kH];           // LSTM cell state
    __shared__ float sgX[16][kNG];         // GRU input-part gate preacts
    __shared__ float sgH[16][kNG];         // GRU hidden-part gate preacts
    __shared__ float sl [16][kNL];         // LSTM summed gate preacts

    const int tid = threadIdx.x;
    const int m0  = blockIdx.x * 16;       // 16 batch rows per wave-tile

    // ---------------- stage weights / biases into LDS (zero padding) --------
    for (int i = tid; i < kNG * kKP; i += 32) {
        int n = i / kKP, k = i % kKP;
        bool ok = (n < kH3) && (k < kH);
        WgX[n][k] = ok ? gWih[n * kH + k] : 0.0f;
        WgH[n][k] = ok ? gWhh[n * kH + k] : 0.0f;
    }
    for (int i = tid; i < kNL * kKP; i += 32) {
        int n = i / kKP, k = i % kKP;
        bool ok = (n < kH4) && (k < kH);
        WlX[n][k] = ok ? lWih[n * kH + k] : 0.0f;
        WlH[n][k] = ok ? lWhh[n * kH + k] : 0.0f;
    }
    for (int i = tid; i < kH3; i += 32) { gbi_s[i] = gbih[i]; gbh_s[i] = gbhh[i]; }
    for (int i = tid; i < kH4; i += 32) { lb_s[i]  = lbih[i] + lbhh[i]; }
    for (int i = tid; i < kH;  i += 32) { fcw_s[i] = fcW[i]; }
    if (tid == 0) fcb_s = fcb[0];
    for (int i = tid; i < 16 * kKP; i += 32) {
        (&xt[0][0])[i] = 0.0f; (&hg[0][0])[i] = 0.0f; (&hl[0][0])[i] = 0.0f;
    }
    for (int i = tid; i < 16 * kH; i += 32) (&cs[0][0])[i] = 0.0f;
    __syncthreads();

    const int lh  = tid & 15;              // A: row M / B: col N / C: col N
    const int kh  = (tid >> 4) << 1;       // 32-bit A/B: hi lanes hold K+2,K+3
    const int hi8 = (tid >> 4) << 3;       // C/D: hi lanes hold rows M+8

    // -------- hoist all WMMA B-fragments (weights) into registers -----------
    v2f bGX[4][kKS], bGH[4][kKS];
    #pragma unroll
    for (int nt = 0; nt < 4; ++nt)
      #pragma unroll
      for (int ks = 0; ks < kKS; ++ks) {
        int r = nt * 16 + lh, c = ks * 4 + kh;
        v2f a; a.x = WgX[r][c]; a.y = WgX[r][c + 1]; bGX[nt][ks] = a;
        v2f b; b.x = WgH[r][c]; b.y = WgH[r][c + 1]; bGH[nt][ks] = b;
      }
    v2f bLX[5][kKS], bLH[5][kKS];
    #pragma unroll
    for (int nt = 0; nt < 5; ++nt)
      #pragma unroll
      for (int ks = 0; ks < kKS; ++ks) {
        int r = nt * 16 + lh, c = ks * 4 + kh;
        v2f a; a.x = WlX[r][c]; a.y = WlX[r][c + 1]; bLX[nt][ks] = a;
        v2f b; b.x = WlH[r][c]; b.y = WlH[r][c + 1]; bLH[nt][ks] = b;
      }

    // ------------------------------ time scan -------------------------------
    for (int t = 0; t < kT; ++t) {
        // stage x[:, t, :] tile into LDS (pads stay zero)
        for (int i = tid; i < 16 * kH; i += 32) {
            int m = i / kH, j = i % kH;
            xt[m][j] = x[((size_t)(m0 + m) * kT + t) * kH + j];
        }
        // prefetch next timestep's rows into cache
        if (t + 1 < kT && tid < 16) {
            __builtin_prefetch(&x[((size_t)(m0 + tid) * kT + (t + 1)) * kH], 0, 3);
        }
        __syncthreads();

        // ---- GRU gate GEMMs: xi = x_t*Wih^T ; hh = h*Whh^T (kept separate) -
        v2f aX[kKS], aH[kKS];
        #pragma unroll
        for (int ks = 0; ks < kKS; ++ks) {
            int c = ks * 4 + kh;
            v2f a; a.x = xt[lh][c]; a.y = xt[lh][c + 1]; aX[ks] = a;
            v2f b; b.x = hg[lh][c]; b.y = hg[lh][c + 1]; aH[ks] = b;
        }
        v8f accX[4], accH[4];
        #pragma unroll
        for (int nt = 0; nt < 4; ++nt) { accX[nt] = (v8f){}; accH[nt] = (v8f){}; }
        #pragma unroll
        for (int nt = 0; nt < 4; ++nt)
          #pragma unroll
          for (int ks = 0; ks < kKS; ++ks) {
            accX[nt] = __builtin_amdgcn_wmma_f32_16x16x4_f32(
                false, aX[ks], false, bGX[nt][ks], (short)0, accX[nt], false, false);
            accH[nt] = __builtin_amdgcn_wmma_f32_16x16x4_f32(
                false, aH[ks], false, bGH[nt][ks], (short)0, accH[nt], false, false);
          }
        #pragma unroll
        for (int nt = 0; nt < 4; ++nt)
          #pragma unroll
          for (int j = 0; j < 8; ++j) {
            sgX[hi8 + j][nt * 16 + lh] = accX[nt][j];
            sgH[hi8 + j][nt * 16 + lh] = accH[nt][j];
          }
        __syncthreads();

        // ---- GRU elementwise: r,z,n -> h_new (h_new is also g_t) ----------
        for (int i = tid; i < 16 * kH; i += 32) {
            int m = i / kH, j = i % kH;
            float r = sigmoidf_((sgX[m][j]          + gbi_s[j])
                              + (sgH[m][j]          + gbh_s[j]));
            float z = sigmoidf_((sgX[m][kH + j]     + gbi_s[kH + j])
                              + (sgH[m][kH + j]     + gbh_s[kH + j]));
            float n = tanhf(    (sgX[m][2*kH + j]   + gbi_s[2*kH + j])
                          + r * (sgH[m][2*kH + j]   + gbh_s[2*kH + j]));
            float hprev = hg[m][j];
            hg[m][j] = (1.0f - z) * n + z * hprev;
        }
        __syncthreads();

        // ---- LSTM gate GEMMs: g = g_t*Wih^T + h*Whh^T (summed accumulator) -
        v2f aG[kKS], aL[kKS];
        #pragma unroll
        for (int ks = 0; ks < kKS; ++ks) {
            int c = ks * 4 + kh;
            v2f a; a.x = hg[lh][c]; a.y = hg[lh][c + 1]; aG[ks] = a;
            v2f b; b.x = hl[lh][c]; b.y = hl[lh][c + 1]; aL[ks] = b;
        }
        v8f accL[5];
        #pragma unroll
        for (int nt = 0; nt < 5; ++nt) accL[nt] = (v8f){};
        #pragma unroll
        for (int nt = 0; nt < 5; ++nt)
          #pragma unroll
          for (int ks = 0; ks < kKS; ++ks) {
            accL[nt] = __builtin_amdgcn_wmma_f32_16x16x4_f32(
                false, aG[ks], false, bLX[nt][ks], (short)0, accL[nt], false, false);
            accL[nt] = __builtin_amdgcn_wmma_f32_16x16x4_f32(
                false, aL[ks], false, bLH[nt][ks], (short)0, accL[nt], false, false);
          }
        #pragma unroll
        for (int nt = 0; nt < 5; ++nt)
          #pragma unroll
          for (int j = 0; j < 8; ++j)
            sl[hi8 + j][nt * 16 + lh] = accL[nt][j];
        __syncthreads();

        // ---- LSTM elementwise: i,f,g,o -> c_new, h_new --------------------
        for (int i = tid; i < 16 * kH; i += 32) {
            int m = i / kH, j = i % kH;
            float ig = sigmoidf_(sl[m][j]          + lb_s[j]);
            float fg = sigmoidf_(sl[m][kH + j]     + lb_s[kH + j]);
            float gg = tanhf(    sl[m][2*kH + j]   + lb_s[2*kH + j]);
            float og = sigmoidf_(sl[m][3*kH + j]   + lb_s[3*kH + j]);
            float cn = fg * cs[m][j] + ig * gg;
            cs[m][j] = cn;
            hl[m][j] = og * tanhf(cn);
        }
        __syncthreads();

        // ---- FC: out[b, t] = h_lstm . fc_W + fc_b -------------------------
        if (tid < 16) {
            float s = fcb_s;
            #pragma unroll
            for (int j = 0; j < kH; ++j) s += hl[tid][j] * fcw_s[j];
            out[(size_t)(m0 + tid) * kT + t] = s;
        }
        __syncthreads();
    }
}

extern "C" void kernel_launch(void* const* d_in, const int* in_sizes, int n_in,
                              void* d_out, int out_size, void* d_ws, size_t ws_size,
                              hipStream_t stream) {
    (void)in_sizes; (void)n_in; (void)out_size; (void)d_ws; (void)ws_size;
    const float* x    = (const float*)d_in[0];
    const float* gWih = (const float*)d_in[1];
    const float* gWhh = (const float*)d_in[2];
    const float* gbih = (const float*)d_in[3];
    const float* gbhh = (const float*)d_in[4];
    const float* lWih = (const float*)d_in[5];
    const float* lWhh = (const float*)d_in[6];
    const float* lbih = (const float*)d_in[7];
    const float* lbhh = (const float*)d_in[8];
    const float* fcW  = (const float*)d_in[9];
    const float* fcb  = (const float*)d_in[10];
    float* out = (float*)d_out;

    dim3 grid(kB / 16);   // 64 independent 16-row batch tiles
    dim3 block(32);       // one wave32 per tile (WMMA needs EXEC all-1s)
    hipLaunchKernelGGL(fused_gru_lstm_fc, grid, block, 0, stream,
                       x, gWih, gWhh, gbih, gbhh,
                       lWih, lWhh, lbih, lbhh, fcW, fcb, out);
}